// InterfaceBoundaryLoss_74259984548555
// MI455X (gfx1250) — compile-verified
//
#include <hip/hip_runtime.h>
#include <hip/hip_bf16.h>

// ---------------------------------------------------------------------------
// InterfaceBoundaryLoss for MI455X (gfx1250, wave32).
//
// Only boundary-ring pixels contribute to the loss (the interface mask is
// mathematically irrelevant: interior gradients are multiplied by bmf==0).
// Roofline: dense evaluation streams 134 MB; this kernel reads the 4 MB bool
// mask (one coalesced dword of 4 mask bytes per thread) plus ~1 MB of sparse
// float gathers at the ring. Cross-lane reduction uses
// V_WMMA_F32_16X16X4_F32 (A = per-lane partials, B = ones):
// D[m,n] = p[m] + p[m+16]; per-lane sum of the 8 C VGPRs + one xor-16
// shuffle yields the exact wave32 sum. Two-stage fixed-order reduction keeps
// the result deterministic across graph replays (no float atomics).
// ---------------------------------------------------------------------------

#define NN 2048
#define NPIX (NN * NN)          // 4,194,304
#define NBATCH 4
#define NBLOCKS 4096            // kernel-1 grid
#define NTHREADS 256            // 8 waves per block
// Each thread owns 4 contiguous pixels -> one uint mask load.
// NBLOCKS*NTHREADS*4 == NPIX.

typedef __attribute__((ext_vector_type(2))) float v2f;
typedef __attribute__((ext_vector_type(8))) float v8f;

// DX as the reference's float32 operand: output(f32) / DX(python float) -> f32
#define DXF ((float)(1.0 / 2047.0))

// Exact wave32 sum via V_WMMA_F32_16X16X4_F32. EXEC must be all-ones
// (all callers keep all 32 lanes active). Result broadcast to all lanes.
__device__ __forceinline__ float wave_sum_wmma(float p) {
    v2f a; a.x = p;    a.y = 0.0f;   // A[m,{0,2}] = p[lane], A[m,{1,3}] = 0
    v2f b; b.x = 1.0f; b.y = 1.0f;   // B = ones(4x16)
    v8f c = {};
    c = __builtin_amdgcn_wmma_f32_16x16x4_f32(
            /*neg_a=*/false, a, /*neg_b=*/false, b,
            /*c_mod=*/(short)0, c, /*reuse_a=*/false, /*reuse_b=*/false);
    // lane L<16 holds D[0..7, L]; lane L>=16 holds D[8..15, L-16].
    float s = c[0] + c[1] + c[2] + c[3] + c[4] + c[5] + c[6] + c[7];
    s += __shfl_xor(s, 16, 32);      // rows 0..7 half + rows 8..15 half
    return s;                        // = sum over all 32 lanes of p
}

// Block (256-thread, 8-wave) reduction of (sum, cnt); valid in thread 0.
__device__ __forceinline__ void block_reduce2(float& sum, float& cnt,
                                              float* ssum, float* scnt) {
    float wsum = wave_sum_wmma(sum);
    float wcnt = wave_sum_wmma(cnt);
    const int wid  = threadIdx.x >> 5;
    const int lane = threadIdx.x & 31;
    if (lane == 0) { ssum[wid] = wsum; scnt[wid] = wcnt; }
    __syncthreads();
    if (threadIdx.x == 0) {
        float bs = 0.0f, bc = 0.0f;
        for (int w = 0; w < (NTHREADS / 32); ++w) { bs += ssum[w]; bc += scnt[w]; }
        sum = bs; cnt = bc;
    }
}

// Per-boundary-pixel contribution across all 4 batches.
__device__ __forceinline__ float pixel_contrib(const float* __restrict__ s1,
                                               const float* __restrict__ s2,
                                               int pid) {
    const int i = pid >> 11;        // row
    const int j = pid & (NN - 1);   // col
    // nx = i*DX - 0.5 > 0  <=>  i >= 1024 (same for j).  roll() wraps mod N.
    const int inb = (i >= NN / 2) ? ((i + 1) & (NN - 1)) : ((i - 1) & (NN - 1));
    const int jnb = (j >= NN / 2) ? ((j + 1) & (NN - 1)) : ((j - 1) & (NN - 1));
    const int px  = inb * NN + j;   // x-neighbor pixel
    const int py  = i * NN + jnb;   // y-neighbor pixel
    float sum = 0.0f;
    #pragma unroll
    for (int b = 0; b < NBATCH; ++b) {
        const int base = b * NPIX;
        const float a1  = s1[base + pid];
        const float a2  = s2[base + pid];
        const float a1x = s1[base + px];
        const float a2x = s2[base + px];
        const float a1y = s1[base + py];
        const float a2y = s2[base + py];
        const float d = a1 - a2;
        sum += d * d;                                // (s1-s2)^2
        float g;
        g = (a1 - a1x) / DXF - 1.0f; sum += g * g;   // (gx1-1)^2
        g = (a1 - a1y) / DXF - 1.0f; sum += g * g;   // (gy1-1)^2
        g = (a2 - a2x) / DXF - 1.0f; sum += g * g;   // (gx2-1)^2
        g = (a2 - a2y) / DXF - 1.0f; sum += g * g;   // (gy2-1)^2
    }
    return sum;
}

// Kernel 1: coalesced mask scan (1 dword = 4 mask bytes per thread) +
// sparse gather at boundary pixels + per-block partial reduction.
__global__ __launch_bounds__(NTHREADS)
void ibl_partial_kernel(const float* __restrict__ s1,
                        const float* __restrict__ s2,
                        const unsigned char* __restrict__ bmask,
                        float* __restrict__ blk_sum,
                        float* __restrict__ blk_cnt) {
    __shared__ float ssum[NTHREADS / 32];
    __shared__ float scnt[NTHREADS / 32];

    const int tid = blockIdx.x * NTHREADS + threadIdx.x;
    const int p0  = tid * 4;                    // 4 contiguous pixels/thread

    float sum = 0.0f, cnt = 0.0f;
    const unsigned int m4 =
        *(const unsigned int*)(bmask + p0);     // coalesced global_load_b32

    if (m4 != 0u) {                             // ~0.15% of threads
        #pragma unroll
        for (int k = 0; k < 4; ++k) {
            if ((m4 >> (8 * k)) & 0xFFu) {
                cnt += 1.0f;
                sum += pixel_contrib(s1, s2, p0 + k);
            }
        }
    }

    block_reduce2(sum, cnt, ssum, scnt);
    if (threadIdx.x == 0) {
        blk_sum[blockIdx.x] = sum;   // every slot overwritten each call
        blk_cnt[blockIdx.x] = cnt;
    }
}

// Kernel 2: deterministic final reduction of NBLOCKS partials -> loss scalar.
__global__ __launch_bounds__(NTHREADS)
void ibl_final_kernel(const float* __restrict__ blk_sum,
                      const float* __restrict__ blk_cnt,
                      float* __restrict__ out) {
    __shared__ float ssum[NTHREADS / 32];
    __shared__ float scnt[NTHREADS / 32];

    float sum = 0.0f, cnt = 0.0f;
    // Fixed-order strided accumulation: deterministic across replays.
    for (int idx = threadIdx.x; idx < NBLOCKS; idx += NTHREADS) {
        sum += blk_sum[idx];
        cnt += blk_cnt[idx];
    }
    block_reduce2(sum, cnt, ssum, scnt);
    if (threadIdx.x == 0) {
        // loss = total / (B * n_bnd), WEIGHT = 1
        out[0] = sum / ((float)NBATCH * cnt);
    }
}

extern "C" void kernel_launch(void* const* d_in, const int* in_sizes, int n_in,
                              void* d_out, int out_size, void* d_ws, size_t ws_size,
                              hipStream_t stream) {
    const float*         s1    = (const float*)d_in[0];          // (4,1,2048,2048) f32
    const float*         s2    = (const float*)d_in[1];          // (4,1,2048,2048) f32
    const unsigned char* bmask = (const unsigned char*)d_in[2];  // (2048,2048) bool
    // d_in[3] = interface mask: provably irrelevant to the loss.

    float* ws      = (float*)d_ws;
    float* blk_sum = ws;             // NBLOCKS floats
    float* blk_cnt = ws + NBLOCKS;   // NBLOCKS floats   (32 KB total)

    ibl_partial_kernel<<<NBLOCKS, NTHREADS, 0, stream>>>(s1, s2, bmask,
                                                         blk_sum, blk_cnt);
    ibl_final_kernel<<<1, NTHREADS, 0, stream>>>(blk_sum, blk_cnt, (float*)d_out);
}